// CPC_69939247448278
// MI455X (gfx1250) — compile-verified
//
#include <hip/hip_runtime.h>

// CPC (wav2vec) fused kernel for MI455X / gfx1250, round 3.
// - fp32 WMMA 16x16x4 (inputs are fp32; workload is bandwidth-bound).
// - proj GEMM fused with InfoNCE logit dot: 603MB proj intermediate never
//   touches memory.
// - W packed as Wt[s][c/4][d][k]: each lane's A-fragment is one b64 load.
// - 64-wide t-tile per workgroup: W L2 traffic = 256 WG x 27MB ~= 6.9GB
//   (4x less than a 16-wide tile); each A-fragment feeds 4 WMMAs.
// - LDS: 192KB x-tile + 2KB reduction buffer (dynamic; CDNA5 WGP has 320KB).

#define Bsz     16
#define Cdim    768
#define Tdim    1024
#define NSTEPS  12
#define NNEG    10
#define OFFS    1
#define NCOPY   11
#define TTILE   64
#define NPAN    (TTILE / 16)         // 4 x 16-column panels
#define PANEL   (Cdim * 16)          // one panel in LDS (floats)

typedef __attribute__((ext_vector_type(2))) float v2f;
typedef __attribute__((ext_vector_type(8))) float v8f;

__global__ void cpc_zero_out(float* out, int n) {
    int i = blockIdx.x * blockDim.x + threadIdx.x;
    if (i < n) out[i] = 0.0f;   // also zeroes int32 labels region (0 bits)
}

// Wt[s][c4][d][k] = W[c4*4+k][d][s]; a lane's (k,k+1) A-pair becomes adjacent.
__global__ void cpc_pack_w(const float* __restrict__ W, float* __restrict__ Wt) {
    int i = blockIdx.x * blockDim.x + threadIdx.x;
    if (i >= NSTEPS * Cdim * Cdim) return;
    int k  = i & 3;
    int d  = (i >> 2) % Cdim;
    int c4 = ((i >> 2) / Cdim) % (Cdim / 4);
    int s  = (i >> 2) / Cdim / (Cdim / 4);
    int c  = c4 * 4 + k;
    Wt[i] = W[(c * Cdim + d) * NSTEPS + s];
}

// Grid: (T/64, B). Block: 256 (8 waves).
// Wave w owns d-subtiles 16*(w + 8j), j=0..5 -> full D=768 per workgroup.
// Each wave accumulates 6 (d) x 4 (t) WMMA tiles.
template <bool PACKED>
__global__ __launch_bounds__(256)
void cpc_fused(const float* __restrict__ x, const float* __restrict__ y,
               const float* __restrict__ Wsrc,
               const float* __restrict__ bias, const int* __restrict__ neg,
               float* __restrict__ out)
{
    extern __shared__ float smem[];
    float* xs  = smem;                   // NPAN panels: [nt][c][16] = 192 KB
    float* red = smem + NPAN * PANEL;    // [8][64] cross-wave reduction

    const int tt   = blockIdx.x;
    const int b    = blockIdx.y;
    const int t0   = tt * TTILE;
    const int tid  = threadIdx.x;
    const int wave = tid >> 5;
    const int lane = tid & 31;
    const int lrow = lane & 15;          // fragment row/col index
    const int kh   = lane >> 4;          // half-wave: K base 2*kh, M base 8*kh

    // Stage x[b, :, t0:t0+64] into LDS (B-operand shared by all 8 waves,
    // reused across all 12 steps and the full C loop).
    const float* xb = x + (size_t)b * Cdim * Tdim + t0;
    for (int i = tid; i < Cdim * TTILE; i += 256) {
        int c = i >> 6, t6 = i & 63;
        xs[(t6 >> 4) * PANEL + c * 16 + (t6 & 15)] = xb[(size_t)c * Tdim + t6];
    }
    __syncthreads();

    for (int s = 0; s < NSTEPS; ++s) {
        const int off = s + OFFS;
        const float* Wi =
            PACKED ? Wsrc + (size_t)s * (Cdim / 4) * Cdim * 4 : Wsrc;

        // Accumulators D[m,n]: m = d row, n = t col; bias folded in at init
        // (the logit dot is linear in proj).
        v8f acc[6][NPAN];
#pragma unroll
        for (int j = 0; j < 6; ++j) {
            const int dm = 16 * (wave + 8 * j) + 8 * kh;
#pragma unroll
            for (int r = 0; r < 8; ++r) {
                float bv = bias[dm + r];
#pragma unroll
                for (int nt = 0; nt < NPAN; ++nt) acc[j][nt][r] = bv;
            }
        }

        // GEMM: proj[d, t] = sum_c W[c,d,s] * x[b,c,t]  (M=d, N=t, K=c)
        for (int c0 = 0; c0 < Cdim; c0 += 4) {
            const int cb = c0 + 2 * kh;
            v2f bf[NPAN];                       // B: 4x16 x-columns per panel
#pragma unroll
            for (int nt = 0; nt < NPAN; ++nt) {
                bf[nt][0] = xs[nt * PANEL + cb * 16 + lrow];
                bf[nt][1] = xs[nt * PANEL + (cb + 1) * 16 + lrow];
            }
#pragma unroll
            for (int j = 0; j < 6; ++j) {
                const int d = 16 * (wave + 8 * j) + lrow;
                v2f af;                         // A: 16x4 W^T rows
                if (PACKED) {
                    // adjacent (k, k+1) pair -> single b64 load
                    af = *(const v2f*)(Wi + ((size_t)(c0 >> 2) * Cdim + d) * 4
                                          + 2 * kh);
                } else {
                    af[0] = Wsrc[((size_t)cb * Cdim + d) * NSTEPS + s];
                    af[1] = Wsrc[((size_t)(cb + 1) * Cdim + d) * NSTEPS + s];
                }
#pragma unroll
                for (int nt = 0; nt < NPAN; ++nt)
                    acc[j][nt] = __builtin_amdgcn_wmma_f32_16x16x4_f32(
                        false, af, false, bf[nt], (short)0, acc[j][nt],
                        false, false);
            }
        }

        // Fused InfoNCE epilogue: p[t,n] = sum_d proj[d,t] * tgt[n,b,d,t+off]
        const long base = (long)Bsz * NCOPY *
                          ((long)s * (Tdim - 1) - (long)s * (s - 1) / 2);

        for (int n = 0; n < NCOPY; ++n) {
#pragma unroll
            for (int nt = 0; nt < NPAN; ++nt) {
                const int tg = t0 + nt * 16 + lrow;
                const bool valid = (tg < Tdim - off);
                const int ti = valid ? (tg + off) : (Tdim - 1);  // clamp loads
                const float* ycol;
                if (n == 0) {
                    ycol = y + (size_t)b * Cdim * Tdim + ti;     // positive
                } else {
                    int idx = neg[(size_t)b * NNEG * Tdim +
                                  (size_t)(n - 1) * Tdim + ti];
                    ycol = y + (size_t)(idx >> 10) * Cdim * Tdim +
                           (idx & (Tdim - 1));
                }
                float ssum = 0.0f;
#pragma unroll
                for (int j = 0; j < 6; ++j) {
                    const int dm = 16 * (wave + 8 * j) + 8 * kh;
#pragma unroll
                    for (int r = 0; r < 8; ++r)
                        ssum += acc[j][nt][r] * ycol[(size_t)(dm + r) * Tdim];
                }
                // combine the two M halves (d and d+8 live in lane^16)
                ssum += __shfl_xor(ssum, 16, 32);
                if (kh == 0) red[wave * TTILE + nt * 16 + lrow] = ssum;
            }
            __syncthreads();
            if (tid < TTILE) {
                float tot = 0.0f;
#pragma unroll
                for (int w = 0; w < 8; ++w) tot += red[w * TTILE + tid];
                const int tg = t0 + tid;
                if (tg < Tdim - off)
                    out[base + ((long)tg * Bsz + b) * NCOPY + n] = tot;
            }
            __syncthreads();
        }
    }
}

extern "C" void kernel_launch(void* const* d_in, const int* in_sizes, int n_in,
                              void* d_out, int out_size, void* d_ws, size_t ws_size,
                              hipStream_t stream) {
    const float* x    = (const float*)d_in[0];
    const float* y    = (const float*)d_in[1];
    const float* W    = (const float*)d_in[2];
    const float* bias = (const float*)d_in[3];
    const int*   neg  = (const int*)d_in[4];
    float* out = (float*)d_out;

    cpc_zero_out<<<(out_size + 255) / 256, 256, 0, stream>>>(out, out_size);

    const size_t wtElems = (size_t)NSTEPS * Cdim * Cdim;
    const size_t shmem   = (size_t)(NPAN * PANEL + 8 * TTILE) * sizeof(float);
    dim3 grid(Tdim / TTILE, Bsz);

    if (ws_size >= wtElems * sizeof(float)) {
        float* Wt = (float*)d_ws;
        cpc_pack_w<<<(int)((wtElems + 255) / 256), 256, 0, stream>>>(W, Wt);
        cpc_fused<true><<<grid, 256, shmem, stream>>>(x, y, Wt, bias, neg, out);
    } else {
        cpc_fused<false><<<grid, 256, shmem, stream>>>(x, y, W, bias, neg, out);
    }
}